// MeshLoss_14439680049450
// MI455X (gfx1250) — compile-verified
//
#include <hip/hip_runtime.h>
#include <hip/hip_bf16.h>

// CDNA5 / gfx1250 wave32 WMMA chamfer + normal loss.
// d2[n,m] = ||x_n||^2 + ( -2 x_n . y_m + ||y_m||^2 )
//           \_per-row_/  \______ one V_WMMA_F32_16X16X4_F32 ______/
// A row = (x0,x1,x2,1), B col = (-2y0,-2y1,-2y2,||y||^2)

typedef __attribute__((ext_vector_type(2))) float v2f;
typedef __attribute__((ext_vector_type(8))) float v8f;

#define CHAMFER_W 1.0f
#define NORMAL_W  0.00016f
#define COS_EPS   1e-6f
#define PAD_BIG   3.0e38f

static __device__ __forceinline__ int pad16(int n) { return (n + 15) & ~15; }

// ---------------------------------------------------------------------------
// Pass 0: pack B-side operands: packX[i] = (-2x0, -2x1, -2x2, ||x||^2).
// Padded tail entries get (0,0,0,PAD_BIG) so D = PAD_BIG there (never the min,
// since running-min init is larger). Also zeros the 8 accumulator floats.
// ---------------------------------------------------------------------------
__global__ void pack_kernel(const float* __restrict__ pp,
                            const float* __restrict__ pg,
                            float4* __restrict__ packP,
                            float4* __restrict__ packG,
                            int N, int M, float* __restrict__ accum) {
  int i = blockIdx.x * blockDim.x + threadIdx.x;
  if (i < 8) accum[i] = 0.0f;
  int Npad = pad16(N), Mpad = pad16(M);
  if (i < Npad) {
    float4 v;
    if (i < N) {
      float x0 = pp[i * 3 + 0], x1 = pp[i * 3 + 1], x2 = pp[i * 3 + 2];
      v = make_float4(-2.0f * x0, -2.0f * x1, -2.0f * x2,
                      x0 * x0 + x1 * x1 + x2 * x2);
    } else {
      v = make_float4(0.0f, 0.0f, 0.0f, PAD_BIG);
    }
    packP[i] = v;
  }
  if (i < Mpad) {
    float4 v;
    if (i < M) {
      float y0 = pg[i * 3 + 0], y1 = pg[i * 3 + 1], y2 = pg[i * 3 + 2];
      v = make_float4(-2.0f * y0, -2.0f * y1, -2.0f * y2,
                      y0 * y0 + y1 * y1 + y2 * y2);
    } else {
      v = make_float4(0.0f, 0.0f, 0.0f, PAD_BIG);
    }
    packG[i] = v;
  }
}

// ---------------------------------------------------------------------------
// One direction of the chamfer loss. One wave (32 lanes) owns a 16-row tile of
// cloud A and scans all column tiles of cloud B with WMMA, keeping per-slot
// running (min, argmin). Then fuses the normal term and atomically accumulates
// sum(cham) and sum(normal_term) into accum[slot*2 + {0,1}].
// ---------------------------------------------------------------------------
__global__ __launch_bounds__(256)
void chamfer_dir_kernel(const float* __restrict__ ptsA,   // [Na,3] raw
                        const float* __restrict__ nrmA,   // [Na,3]
                        const float* __restrict__ nrmB,   // [Mb,3]
                        const float* __restrict__ packB,  // [pad16(Mb)*4]
                        int Na, int Mb,
                        float* __restrict__ accum, int slot) {
  const int lane = threadIdx.x & 31;
  const int waveId = (blockIdx.x * blockDim.x + threadIdx.x) >> 5;
  const int rb = waveId * 16;
  if (rb >= Na) return;  // uniform per wave

  const int l16 = lane & 15;
  const int hi  = (lane >> 4) & 1;  // lanes 16-31 carry K=2,3 / rows 8-15

  // ---- A operand: lane<16 -> (x0,x1) of row rb+l16 ; lane>=16 -> (x2, 1)
  int arow = rb + l16;
  if (arow >= Na) arow = Na - 1;  // Na is a multiple of 16 in practice
  v2f A;
  if (!hi) {
    A.x = ptsA[arow * 3 + 0];
    A.y = ptsA[arow * 3 + 1];
  } else {
    A.x = ptsA[arow * 3 + 2];
    A.y = 1.0f;
  }

  float minv[8];
  int   mini[8];
#pragma unroll
  for (int v = 0; v < 8; ++v) { minv[v] = 3.4e38f; mini[v] = 0; }

  const v8f czero = {};
  const int numCT = (Mb + 15) >> 4;
  const v2f* packB2 = (const v2f*)packB;  // 2 halves (xy | zw) per packed point

  for (int ct = 0; ct < numCT; ++ct) {
    const int col = ct * 16 + l16;
    // lane<16: (-2y0,-2y1) of col ; lane>=16: (-2y2, ||y||^2) of col
    v2f B = packB2[col * 2 + hi];
    // D = A x B : D[n,m] = -2 x_n . y_m + ||y_m||^2
    v8f D = __builtin_amdgcn_wmma_f32_16x16x4_f32(
        false, A, false, B, (short)0, czero, false, false);
#pragma unroll
    for (int v = 0; v < 8; ++v) {
      bool lt = D[v] < minv[v];
      minv[v] = lt ? D[v] : minv[v];
      mini[v] = lt ? col  : mini[v];
    }
  }

  // ---- reduce each slot across the 16 lanes of each half-wave
#pragma unroll
  for (int v = 0; v < 8; ++v) {
#pragma unroll
    for (int off = 1; off < 16; off <<= 1) {
      float ov = __shfl_xor(minv[v], off, 32);
      int   oi = __shfl_xor(mini[v], off, 32);
      if (ov < minv[v]) { minv[v] = ov; mini[v] = oi; }
    }
  }

  // ---- lanes 0-7 own rows rb+0..7 (slots 0..7, low half);
  //      lanes 16-23 own rows rb+8..15 (slots 0..7, high half)
  float chamC = 0.0f, normC = 0.0f;
  if (l16 < 8) {
    int r = rb + l16 + (hi ? 8 : 0);
    if (r < Na) {
      float mval = minv[l16];
      int   midx = mini[l16];
      float x0 = ptsA[r * 3 + 0], x1 = ptsA[r * 3 + 1], x2 = ptsA[r * 3 + 2];
      chamC = mval + (x0 * x0 + x1 * x1 + x2 * x2);  // add back ||x||^2
      // normal term: 1 - |cos(nA_r, nB_midx)|
      float a0 = nrmA[r * 3 + 0], a1 = nrmA[r * 3 + 1], a2 = nrmA[r * 3 + 2];
      float b0 = nrmB[midx * 3 + 0], b1 = nrmB[midx * 3 + 1], b2 = nrmB[midx * 3 + 2];
      float num = a0 * b0 + a1 * b1 + a2 * b2;
      float den = sqrtf((a0 * a0 + a1 * a1 + a2 * a2) *
                        (b0 * b0 + b1 * b1 + b2 * b2));
      den = fmaxf(den, COS_EPS);
      normC = 1.0f - fabsf(num / den);
    }
  }

  // ---- full-wave sum then one atomic per wave
#pragma unroll
  for (int off = 1; off < 32; off <<= 1) {
    chamC += __shfl_xor(chamC, off, 32);
    normC += __shfl_xor(normC, off, 32);
  }
  if (lane == 0) {
    atomicAdd(&accum[slot * 2 + 0], chamC);
    atomicAdd(&accum[slot * 2 + 1], normC);
  }
}

// ---------------------------------------------------------------------------
// Combine: total = W_c*(sum_x/N + sum_y/M) + W_n*(nsum_x/N + nsum_y/M)
// ---------------------------------------------------------------------------
__global__ void finalize_kernel(const float* __restrict__ accum,
                                float* __restrict__ out, int N, int M) {
  if (threadIdx.x == 0 && blockIdx.x == 0) {
    float cham = accum[0] / (float)N + accum[2] / (float)M;
    float nrm  = accum[1] / (float)N + accum[3] / (float)M;
    out[0] = CHAMFER_W * cham + NORMAL_W * nrm;
  }
}

extern "C" void kernel_launch(void* const* d_in, const int* in_sizes, int n_in,
                              void* d_out, int out_size, void* d_ws, size_t ws_size,
                              hipStream_t stream) {
  const float* pp = (const float*)d_in[0];  // points_pred  [1,N,3]
  const float* np = (const float*)d_in[1];  // normals_pred [1,N,3]
  const float* pg = (const float*)d_in[2];  // points_gt    [1,M,3]
  const float* ng = (const float*)d_in[3];  // normals_gt   [1,M,3]
  const int N = in_sizes[0] / 3;
  const int M = in_sizes[2] / 3;
  const int Npad = (N + 15) & ~15;
  const int Mpad = (M + 15) & ~15;

  // workspace: packP [Npad*4 f32] | packG [Mpad*4 f32] | accum [8 f32]
  float* packP = (float*)d_ws;
  float* packG = packP + (size_t)Npad * 4;
  float* accum = packG + (size_t)Mpad * 4;

  const int mx = (Npad > Mpad ? Npad : Mpad);
  pack_kernel<<<(mx + 255) / 256, 256, 0, stream>>>(
      pp, pg, (float4*)packP, (float4*)packG, N, M, accum);

  // pred -> gt : rows = pred, cols = gt, gather gt normals
  {
    int tiles = Npad / 16;                 // one wave per tile
    int blocks = (tiles * 32 + 255) / 256;
    chamfer_dir_kernel<<<blocks, 256, 0, stream>>>(pp, np, ng, packG, N, M,
                                                   accum, 0);
  }
  // gt -> pred : rows = gt, cols = pred, gather pred normals
  {
    int tiles = Mpad / 16;
    int blocks = (tiles * 32 + 255) / 256;
    chamfer_dir_kernel<<<blocks, 256, 0, stream>>>(pg, ng, np, packP, M, N,
                                                   accum, 1);
  }
  finalize_kernel<<<1, 32, 0, stream>>>(accum, (float*)d_out, N, M);
}